// KGAT_43817256354272
// MI455X (gfx1250) — compile-verified
//
#include <hip/hip_runtime.h>
#include <hip/hip_bf16.h>

typedef __attribute__((ext_vector_type(2))) float v2f;
typedef __attribute__((ext_vector_type(8))) float v8f;

// ---------------------------------------------------------------------------
// Utility: zero-fill a float buffer (graph-capture safe, no memset)
// ---------------------------------------------------------------------------
__global__ void kgat_zero_kernel(float* __restrict__ p, long long n) {
    long long i = (long long)blockIdx.x * blockDim.x + threadIdx.x;
    long long stride = (long long)gridDim.x * blockDim.x;
    for (; i < n; i += stride) p[i] = 0.0f;
}

// ---------------------------------------------------------------------------
// Copy ego embeddings into output columns [0, 64)
// ---------------------------------------------------------------------------
__global__ void kgat_copy_ego_kernel(const float* __restrict__ ego,
                                     float* __restrict__ out,
                                     int n, int d, int ostride) {
    long long total = (long long)n * d;
    long long i = (long long)blockIdx.x * blockDim.x + threadIdx.x;
    long long stride = (long long)gridDim.x * blockDim.x;
    for (; i < total; i += stride) {
        int r = (int)(i / d);
        int c = (int)(i % d);
        out[(long long)r * ostride + c] = ego[i];
    }
}

// ---------------------------------------------------------------------------
// SpMM: side[row[e], :] += val[e] * x[col[e], :]   (COO scatter-add)
// One thread per (edge, 4-float chunk). x fits in the 192MB L2, so gathers
// are L2-resident; scatter uses global_atomic_add_f32.
// ---------------------------------------------------------------------------
template <int D>
__global__ __launch_bounds__(256)
void kgat_spmm_kernel(const int* __restrict__ erow, const int* __restrict__ ecol,
                      const float* __restrict__ eval, const float* __restrict__ x,
                      float* __restrict__ side, long long nwork) {
    constexpr int CH = D / 4;  // float4 chunks per edge
    long long idx = (long long)blockIdx.x * blockDim.x + threadIdx.x;
    if (idx >= nwork) return;
    int e   = (int)(idx / CH);
    int sub = (int)(idx % CH);
    float v = eval[e];
    int c = ecol[e];
    int r = erow[e];
    const float4 xv = *(const float4*)(x + (long long)c * D + sub * 4);
    float* dst = side + (long long)r * D + sub * 4;
    atomicAdd(dst + 0, v * xv.x);
    atomicAdd(dst + 1, v * xv.y);
    atomicAdd(dst + 2, v * xv.z);
    atomicAdd(dst + 3, v * xv.w);
}

// ---------------------------------------------------------------------------
// Fused transform (the dense part of one KGAT layer), WMMA fp32 path:
//   h1 = x + side;  h2 = x * side
//   xout = leaky(h1 @ W1 + b1) + leaky(h2 @ W2 + b2)
//
// Per-wave 16-row M tile. V_WMMA_F32_16X16X4_F32 layouts (ISA 7.12.2):
//   A (16x4): lanes 0-15 -> M=lane, K={0,1} in VGPR{0,1}; lanes 16-31 -> K={2,3}
//   B (4x16): lanes 0-15 -> N=lane, K={0,1}; lanes 16-31 -> K={2,3}
//   C/D (16x16): VGPR r: lanes 0-15 -> M=r, N=lane; lanes 16-31 -> M=r+8
// ---------------------------------------------------------------------------
template <int DIN, int DOUT>
__global__ __launch_bounds__(256)
void kgat_transform_kernel(const float* __restrict__ x,
                           const float* __restrict__ side,
                           const float* __restrict__ W1, const float* __restrict__ b1,
                           const float* __restrict__ W2, const float* __restrict__ b2,
                           float* __restrict__ xout, int n) {
    constexpr int KSTEPS = DIN / 4;
    constexpr int NTILES = DOUT / 16;

    const int lane  = threadIdx.x & 31;
    const int wave  = threadIdx.x >> 5;                 // 8 waves per block
    const int mBase = (blockIdx.x * 8 + wave) * 16;     // 16 rows per wave
    if (mBase >= n) return;                             // wave-uniform exit

    // A-fragment lane mapping
    int rowA = mBase + (lane & 15);
    if (rowA > n - 1) rowA = n - 1;                     // clamp reads (writes guarded)
    const int kOff = (lane < 16) ? 0 : 2;

    // Preload A fragments for both branches (reused across all N tiles)
    v2f a1[KSTEPS], a2[KSTEPS];
#pragma unroll
    for (int k = 0; k < KSTEPS; ++k) {
        const int kk = k * 4 + kOff;
        const float xv0 = x[(long long)rowA * DIN + kk];
        const float xv1 = x[(long long)rowA * DIN + kk + 1];
        const float sv0 = side[(long long)rowA * DIN + kk];
        const float sv1 = side[(long long)rowA * DIN + kk + 1];
        a1[k] = v2f{xv0 + sv0, xv1 + sv1};   // sum branch
        a2[k] = v2f{xv0 * sv0, xv1 * sv1};   // bi (hadamard) branch
    }

    const int nCol = lane & 15;
#pragma unroll
    for (int nt = 0; nt < NTILES; ++nt) {
        const int col = nt * 16 + nCol;
        v8f acc1 = {};
        v8f acc2 = {};
#pragma unroll
        for (int k = 0; k < KSTEPS; ++k) {
            const int kk = k * 4 + kOff;
            v2f bw1 = v2f{W1[(long long)kk * DOUT + col], W1[(long long)(kk + 1) * DOUT + col]};
            v2f bw2 = v2f{W2[(long long)kk * DOUT + col], W2[(long long)(kk + 1) * DOUT + col]};
            acc1 = __builtin_amdgcn_wmma_f32_16x16x4_f32(
                false, a1[k], false, bw1, (short)0, acc1, false, false);
            acc2 = __builtin_amdgcn_wmma_f32_16x16x4_f32(
                false, a2[k], false, bw2, (short)0, acc2, false, false);
        }

        // Epilogue: bias + LeakyReLU(0.01) on both branches, then sum
        const float bias1 = b1[col];
        const float bias2 = b2[col];
        const int rOff = (lane < 16) ? 0 : 8;
#pragma unroll
        for (int r = 0; r < 8; ++r) {
            const int row = mBase + r + rOff;
            if (row < n) {
                float v1 = acc1[r] + bias1;
                v1 = (v1 >= 0.0f) ? v1 : 0.01f * v1;
                float v2 = acc2[r] + bias2;
                v2 = (v2 >= 0.0f) ? v2 : 0.01f * v2;
                xout[(long long)row * DOUT + col] = v1 + v2;
            }
        }
    }
}

// ---------------------------------------------------------------------------
// Row-wise L2 normalize + write into output column slice. One wave per row.
// ---------------------------------------------------------------------------
template <int D>
__global__ __launch_bounds__(256)
void kgat_l2norm_kernel(const float* __restrict__ xin, float* __restrict__ out,
                        int ostride, int colOff, int n) {
    const int lane = threadIdx.x & 31;
    const int row  = blockIdx.x * (blockDim.x >> 5) + (threadIdx.x >> 5);
    if (row >= n) return;

    float ss = 0.0f;
#pragma unroll
    for (int c = lane; c < D; c += 32) {
        const float v = xin[(long long)row * D + c];
        ss += v * v;
    }
#pragma unroll
    for (int off = 16; off > 0; off >>= 1) ss += __shfl_xor(ss, off, 32);

    const float norm = sqrtf(ss);
    const float inv  = 1.0f / fmaxf(norm, 1e-12f);
#pragma unroll
    for (int c = lane; c < D; c += 32) {
        out[(long long)row * ostride + colOff + c] =
            xin[(long long)row * D + c] * inv;
    }
}

// ---------------------------------------------------------------------------
// Launch: 3 KGAT layers. Workspace: side (N*64) | bufA (N*64) | bufB (N*64)
// ---------------------------------------------------------------------------
extern "C" void kernel_launch(void* const* d_in, const int* in_sizes, int n_in,
                              void* d_out, int out_size, void* d_ws, size_t ws_size,
                              hipStream_t stream) {
    const float* ego  = (const float*)d_in[0];
    const int*   erow = (const int*)d_in[1];
    const int*   ecol = (const int*)d_in[2];
    const float* eval = (const float*)d_in[3];
    const float* W1_0 = (const float*)d_in[4];
    const float* b1_0 = (const float*)d_in[5];
    const float* W2_0 = (const float*)d_in[6];
    const float* b2_0 = (const float*)d_in[7];
    const float* W1_1 = (const float*)d_in[8];
    const float* b1_1 = (const float*)d_in[9];
    const float* W2_1 = (const float*)d_in[10];
    const float* b2_1 = (const float*)d_in[11];
    const float* W1_2 = (const float*)d_in[12];
    const float* b1_2 = (const float*)d_in[13];
    const float* W2_2 = (const float*)d_in[14];
    const float* b2_2 = (const float*)d_in[15];

    const int N = in_sizes[0] / 64;      // 160000
    const long long E = in_sizes[1];     // 5,120,000
    const int OSTRIDE = 64 + 64 + 32 + 16;  // 176
    float* out = (float*)d_out;

    float* side = (float*)d_ws;                          // N*64
    float* bufA = side + (long long)N * 64;              // N*64
    float* bufB = bufA + (long long)N * 64;              // N*64

    const int ZB = 1024;  // blocks for fill/copy grid-stride kernels

    // Output columns [0,64): raw ego
    kgat_copy_ego_kernel<<<ZB, 256, 0, stream>>>(ego, out, N, 64, OSTRIDE);

    // ---------------- Layer 0: 64 -> 64 ----------------
    {
        const long long nside = (long long)N * 64;
        kgat_zero_kernel<<<ZB, 256, 0, stream>>>(side, nside);
        const long long nwork = E * (64 / 4);
        kgat_spmm_kernel<64><<<(int)((nwork + 255) / 256), 256, 0, stream>>>(
            erow, ecol, eval, ego, side, nwork);
        kgat_transform_kernel<64, 64><<<(N + 127) / 128, 256, 0, stream>>>(
            ego, side, W1_0, b1_0, W2_0, b2_0, bufA, N);
        kgat_l2norm_kernel<64><<<(N + 7) / 8, 256, 0, stream>>>(
            bufA, out, OSTRIDE, 64, N);
    }

    // ---------------- Layer 1: 64 -> 32 ----------------
    {
        const long long nside = (long long)N * 64;
        kgat_zero_kernel<<<ZB, 256, 0, stream>>>(side, nside);
        const long long nwork = E * (64 / 4);
        kgat_spmm_kernel<64><<<(int)((nwork + 255) / 256), 256, 0, stream>>>(
            erow, ecol, eval, bufA, side, nwork);
        kgat_transform_kernel<64, 32><<<(N + 127) / 128, 256, 0, stream>>>(
            bufA, side, W1_1, b1_1, W2_1, b2_1, bufB, N);
        kgat_l2norm_kernel<32><<<(N + 7) / 8, 256, 0, stream>>>(
            bufB, out, OSTRIDE, 128, N);
    }

    // ---------------- Layer 2: 32 -> 16 ----------------
    {
        const long long nside = (long long)N * 32;
        kgat_zero_kernel<<<ZB, 256, 0, stream>>>(side, nside);
        const long long nwork = E * (32 / 4);
        kgat_spmm_kernel<32><<<(int)((nwork + 255) / 256), 256, 0, stream>>>(
            erow, ecol, eval, bufB, side, nwork);
        kgat_transform_kernel<32, 16><<<(N + 127) / 128, 256, 0, stream>>>(
            bufB, side, W1_2, b1_2, W2_2, b2_2, bufA, N);
        kgat_l2norm_kernel<16><<<(N + 7) / 8, 256, 0, stream>>>(
            bufA, out, OSTRIDE, 160, N);
    }
}